// ASREnv_20899310863100
// MI455X (gfx1250) — compile-verified
//
#include <hip/hip_runtime.h>

typedef float v2f __attribute__((ext_vector_type(2)));
typedef float v8f __attribute__((ext_vector_type(8)));

#define TIME_STEPS 4096
#define TILE 256                                // 16x16 elements per tile
#define TILES_PER_PATH (TIME_STEPS / TILE)      // 16
#define WAVES_PER_BLOCK 8

__device__ __forceinline__ float rlane(float x, int l) {
    return __uint_as_float(__builtin_amdgcn_readlane(__float_as_uint(x), l));
}

// Within-tile scan: Y = X * U  (U = upper-triangular ones, incl. diagonal),
// computed as 4 chained V_WMMA_F32_16X16X4_F32 k-slices. a[j] holds the A
// fragment for columns 4j..4j+3 of X; B fragment of U is synthesized (and
// gets hoisted to loop-invariant registers by the compiler).
__device__ __forceinline__ v8f tile_scan(const v2f a[4], int halfSel, int n) {
    v8f c = {0.f, 0.f, 0.f, 0.f, 0.f, 0.f, 0.f, 0.f};
#pragma unroll
    for (int j = 0; j < 4; ++j) {
        // B 4x16 fragment: reg g, lane l -> element (K = 4j + g + 2*halfSel, N = n)
        const int k0 = 4 * j + 0 + 2 * halfSel;
        const int k1 = 4 * j + 1 + 2 * halfSel;
        v2f b;
        b.x = (k0 <= n) ? 1.0f : 0.0f;
        b.y = (k1 <= n) ? 1.0f : 0.0f;
        c = __builtin_amdgcn_wmma_f32_16x16x4_f32(
                /*neg_a=*/false, a[j], /*neg_b=*/false, b,
                /*c_mod=*/(short)0, c, /*reuse_a=*/false, /*reuse_b=*/false);
    }
    return c;
}

// Add cross-row carries (exclusive scan of the 16 row totals, found in column
// N=15 of the D fragment) plus the running inter-tile carry. Row totals are
// pulled from fixed lanes with v_readlane (uniform values) -> no LDS traffic,
// the 16-element exclusive scan is 15 dependent uniform adds.
// Returns the updated running carry (old carry + tile total).
__device__ __forceinline__ float cross_row_fix(v8f& acc, int halfSel,
                                               float runCarry) {
    float t[16];
#pragma unroll
    for (int d = 0; d < 8; ++d) {
        t[d]     = rlane(acc[d], 15);   // Y[d, 15]
        t[d + 8] = rlane(acc[d], 31);   // Y[d+8, 15]
    }
    float e[16];                        // e[i] = runCarry + sum_{i'<i} t[i']
    e[0] = runCarry;
#pragma unroll
    for (int i = 1; i < 16; ++i) e[i] = e[i - 1] + t[i - 1];
#pragma unroll
    for (int d = 0; d < 8; ++d) {
        acc[d] += halfSel ? e[d + 8] : e[d];   // carry for M = d + 8*halfSel
    }
    return e[15] + t[15];
}

__global__ void __launch_bounds__(256)
vwap_scan_kernel(const float* __restrict__ S, const float* __restrict__ V,
                 float* __restrict__ out) {
    const int lane    = threadIdx.x & 31;
    const int path    = blockIdx.x * WAVES_PER_BLOCK + (threadIdx.x >> 5);
    const int halfSel = lane >> 4;     // 0: lanes 0-15, 1: lanes 16-31
    const int n       = lane & 15;

    const size_t rowBase = (size_t)path * TIME_STEPS;
    const float* sRow = S + rowBase;
    const float* vRow = V + rowBase;
    float*       oRow = out + rowBase;

    float runPV = 0.f;
    float runV  = 0.f;

    for (int tile = 0; tile < TILES_PER_PATH; ++tile) {
        const int tbase = tile * TILE;

        if (tile + 1 < TILES_PER_PATH) {
            // Each lane prefetches 32B of the next 1KB tile -> whole tile covered.
            __builtin_prefetch(sRow + tbase + TILE + lane * 8, 0, 0);
            __builtin_prefetch(vRow + tbase + TILE + lane * 8, 0, 0);
        }

        // Load A fragments: slice j, reg g -> element (row = n, col = 4j+g+2*halfSel)
        // memory linear index = 16*row + col (row-major 16x16 over 256 floats).
        v2f as[4], av[4], apv[4];
#pragma unroll
        for (int j = 0; j < 4; ++j) {
            const int off = tbase + 16 * n + 4 * j + 2 * halfSel;  // 8B-aligned
            as[j] = *(const v2f*)(sRow + off);
            av[j] = *(const v2f*)(vRow + off);
            apv[j].x = as[j].x * av[j].x;
            apv[j].y = as[j].y * av[j].y;
        }

        v8f cpv = tile_scan(apv, halfSel, n);   // scan of S*V within tile
        v8f cv  = tile_scan(av,  halfSel, n);   // scan of V   within tile

        runPV = cross_row_fix(cpv, halfSel, runPV);
        runV  = cross_row_fix(cv,  halfSel, runV);

        // D layout: VGPR d, element (M = d + 8*halfSel, N = n) -> 16*M + N.
#pragma unroll
        for (int d = 0; d < 8; ++d) {
            float val = cpv[d] / (cv[d] + 1e-8f);
            if (tile == 0 && d == 0 && lane == 0) {
                val = as[0].x;   // reference: vwap[:,0] = S[:,0] exactly
            }
            oRow[tbase + 16 * d + 128 * halfSel + n] = val;
        }
    }
}

extern "C" void kernel_launch(void* const* d_in, const int* in_sizes, int n_in,
                              void* d_out, int out_size, void* d_ws, size_t ws_size,
                              hipStream_t stream) {
    const float* S = (const float*)d_in[0];
    const float* V = (const float*)d_in[1];
    float* out = (float*)d_out;

    const int num_paths = in_sizes[0] / TIME_STEPS;   // 16384
    const int blocks = num_paths / WAVES_PER_BLOCK;   // 2048 blocks of 8 waves

    vwap_scan_kernel<<<blocks, 256, 0, stream>>>(S, V, out);
}